// FourierImageEventFusion_48765058679349
// MI455X (gfx1250) — compile-verified
//
#include <hip/hip_runtime.h>
#include <hip/hip_bf16.h>
#include <math.h>

// ---------------------------------------------------------------------------
// FourierImageEventFusion for MI455X (gfx1250, wave32).
// f32 end-to-end (memory-bound: ~12 GFLOP vs ~0.5 GB min traffic @ 23.3 TB/s).
// GEMMs (the three 1x1 convs) use V_WMMA_F32_16X16X4_F32.
// ---------------------------------------------------------------------------

typedef float v2f __attribute__((ext_vector_type(2)));
typedef float v8f __attribute__((ext_vector_type(8)));

#define RSQ2 0.70710678118654752440f
#define HWIMG 65536      // 256*256
#define BATCH 4
#define CIN   64
#define HID2  128

// ------------------------------ complex helpers -----------------------------
struct cpx { float re, im; };
__device__ __forceinline__ cpx cadd(cpx a, cpx b) { return {a.re + b.re, a.im + b.im}; }
__device__ __forceinline__ cpx csub(cpx a, cpx b) { return {a.re - b.re, a.im - b.im}; }
__device__ __forceinline__ cpx cmul(cpx a, cpx b) {
    return {a.re * b.re - a.im * b.im, a.re * b.im + a.im * b.re};
}
__device__ __forceinline__ cpx rot_mi(cpx a) { return { a.im, -a.re}; }  // * -i
__device__ __forceinline__ cpx rot_pi(cpx a) { return {-a.im,  a.re}; }  // * +i

// 8-point complex FFT, radix-2 DIT, hardcoded twiddles. FWD: e^{-i}, else e^{+i}
// (inverse variant is UNSCALED; 1/64 folded into the final row inverse).
template <bool FWD>
__device__ __forceinline__ void fft8c(cpx v[8]) {
    cpx t0 = cadd(v[0], v[4]), t1 = csub(v[0], v[4]);
    cpx t2 = cadd(v[2], v[6]), t3 = csub(v[2], v[6]);
    cpx t4 = cadd(v[1], v[5]), t5 = csub(v[1], v[5]);
    cpx t6 = cadd(v[3], v[7]), t7 = csub(v[3], v[7]);

    cpx r3 = FWD ? rot_mi(t3) : rot_pi(t3);
    cpx r7 = FWD ? rot_mi(t7) : rot_pi(t7);
    cpx u0 = cadd(t0, t2), u2 = csub(t0, t2);
    cpx u1 = cadd(t1, r3), u3 = csub(t1, r3);
    cpx u4 = cadd(t4, t6), u6 = csub(t4, t6);
    cpx u5 = cadd(t5, r7), u7 = csub(t5, r7);

    const cpx w1 = FWD ? cpx{ RSQ2, -RSQ2} : cpx{ RSQ2, RSQ2};
    const cpx w3 = FWD ? cpx{-RSQ2, -RSQ2} : cpx{-RSQ2, RSQ2};
    cpx s4 = u4;
    cpx s5 = cmul(w1, u5);
    cpx s6 = FWD ? rot_mi(u6) : rot_pi(u6);
    cpx s7 = cmul(w3, u7);

    v[0] = cadd(u0, s4); v[4] = csub(u0, s4);
    v[1] = cadd(u1, s5); v[5] = csub(u1, s5);
    v[2] = cadd(u2, s6); v[6] = csub(u2, s6);
    v[3] = cadd(u3, s7); v[7] = csub(u3, s7);
}

// real 8 -> onesided 5 complex (forward, e^{-i})
__device__ __forceinline__ void rfft8(const float x[8], cpx X[5]) {
    float p = x[0] - x[4];
    float q = x[2] - x[6];
    float r = x[1] - x[3] - x[5] + x[7];
    float s = -x[1] - x[3] + x[5] + x[7];
    X[0] = { x[0]+x[1]+x[2]+x[3]+x[4]+x[5]+x[6]+x[7], 0.f };
    X[4] = { x[0]-x[1]+x[2]-x[3]+x[4]-x[5]+x[6]-x[7], 0.f };
    X[2] = { x[0]-x[2]+x[4]-x[6], -x[1]+x[3]-x[5]+x[7] };
    X[1] = { p + RSQ2 * r, -q + RSQ2 * s };
    X[3] = { p - RSQ2 * r,  q + RSQ2 * s };
}

// onesided 5 complex -> real 8, includes the full 2-D 1/64 scale
__device__ __forceinline__ void irfft8_scaled(const cpx X[5], float o[8]) {
    const float CT[8] = {1.f,  RSQ2, 0.f, -RSQ2, -1.f, -RSQ2,  0.f,  RSQ2};
    const float ST[8] = {0.f,  RSQ2, 1.f,  RSQ2,  0.f, -RSQ2, -1.f, -RSQ2};
#pragma unroll
    for (int n = 0; n < 8; ++n) {
        float v = X[0].re + X[4].re * CT[(4 * n) & 7];
        v += 2.f * (X[1].re * CT[n & 7]       - X[1].im * ST[n & 7]);
        v += 2.f * (X[2].re * CT[(2 * n) & 7] - X[2].im * ST[(2 * n) & 7]);
        v += 2.f * (X[3].re * CT[(3 * n) & 7] - X[3].im * ST[(3 * n) & 7]);
        o[n] = v * (1.0f / 64.0f);
    }
}

// ---------------------------------------------------------------------------
// Kernel: 1x1 conv as GEMM with V_WMMA_F32_16X16X4_F32.
//   out[b, m, p] = sum_c wt[m, c] * in[b, c, p]
// One wave owns a 16(out-ch) x 16(pixel) tile; K advances 4 per WMMA.
// OCT = OC/16 M-tiles, NT pixel-tiles per block, OCT*NT == 8 waves (256 thr).
// f32 WMMA VGPR layout (ISA 7.12.2):
//   A (16x4): lanes 0-15 -> M=lane, v0={K=0}, v1={K=1}; lanes 16-31 K+=2
//   B (4x16): v0: lanes0-15 K=0 / lanes16-31 K=1 ; v1: K=2 / K=3, N=lane&15
//   C/D     : v[r] = row (r + 8*(lane>=16)), col = lane&15
// ---------------------------------------------------------------------------
template <int OC, int IC, int OCT, int NT>
__global__ __launch_bounds__(256) void conv1x1_wmma(const float* __restrict__ in,
                                                    const float* __restrict__ wt,
                                                    float* __restrict__ out) {
    const int lane = threadIdx.x & 31;
    const int wave = threadIdx.x >> 5;
    const int hi   = lane >> 4;        // 0 | 1 (lane half)
    const int lm   = lane & 15;

    const int m0 = (wave % OCT) * 16;
    const int n0 = (blockIdx.x * NT + (wave / OCT)) * 16;
    const int b  = blockIdx.y;

    const float* __restrict__ inb  = in  + (size_t)b * IC * HWIMG;
    float* __restrict__       outb = out + (size_t)b * OC * HWIMG;

    v8f acc = {0.f, 0.f, 0.f, 0.f, 0.f, 0.f, 0.f, 0.f};

#pragma unroll
    for (int k0 = 0; k0 < IC; k0 += 4) {
        v2f a, bm;
        const int ka = k0 + hi * 2;
        a.x = wt[(m0 + lm) * IC + ka];
        a.y = wt[(m0 + lm) * IC + ka + 1];
        bm.x = inb[(size_t)(k0 + hi)     * HWIMG + n0 + lm];
        bm.y = inb[(size_t)(k0 + 2 + hi) * HWIMG + n0 + lm];
        acc = __builtin_amdgcn_wmma_f32_16x16x4_f32(
            /*neg_a=*/false, a, /*neg_b=*/false, bm,
            /*c_mod=*/(short)0, acc, /*reuse_a=*/false, /*reuse_b=*/false);
    }

#pragma unroll
    for (int r = 0; r < 8; ++r)
        outb[(size_t)(m0 + r + hi * 8) * HWIMG + n0 + lm] = acc[r];
}

// ---------------------------------------------------------------------------
// Kernel: patchwise 8x8 rfft2 / complex fuse / irfft2.
// One thread per (batch, channel, patch). Writes x in place over img_ region
// (each thread owns its patch -> race free).
//   x = irfft2( (rfft2(img_) + 1) * filter * rfft2(event_) )
// ---------------------------------------------------------------------------
__global__ __launch_bounds__(256) void fft_fuse(const float* __restrict__ xi,
                                                const float* __restrict__ xe,
                                                const float* __restrict__ filt,
                                                float* __restrict__ xo) {
    int t = blockIdx.x * blockDim.x + threadIdx.x;
    const int px = t & 31;  t >>= 5;
    const int py = t & 31;  t >>= 5;
    const int ch = t & 127; t >>= 7;
    const int b  = t;

    const size_t base = ((size_t)(b * HID2 + ch)) * HWIMG + (size_t)py * 8 * 256 + px * 8;

    // ---- event forward FFT, scaled by filter ----
    cpx E[8][5];
#pragma unroll
    for (int r = 0; r < 8; ++r) {
        const float4* rp = reinterpret_cast<const float4*>(xe + base + (size_t)r * 256);
        float4 lo = rp[0], hiq = rp[1];
        float row[8] = {lo.x, lo.y, lo.z, lo.w, hiq.x, hiq.y, hiq.z, hiq.w};
        rfft8(row, E[r]);
    }
#pragma unroll
    for (int k = 0; k < 5; ++k) {
        cpx v[8];
#pragma unroll
        for (int r = 0; r < 8; ++r) v[r] = E[r][k];
        fft8c<true>(v);
#pragma unroll
        for (int r = 0; r < 8; ++r) E[r][k] = v[r];
    }
#pragma unroll
    for (int r = 0; r < 8; ++r)
#pragma unroll
        for (int k = 0; k < 5; ++k) {
            float f = filt[ch * 40 + r * 5 + k];
            E[r][k].re *= f;
            E[r][k].im *= f;
        }

    // ---- img forward FFT ----
    cpx I[8][5];
#pragma unroll
    for (int r = 0; r < 8; ++r) {
        const float4* rp = reinterpret_cast<const float4*>(xi + base + (size_t)r * 256);
        float4 lo = rp[0], hiq = rp[1];
        float row[8] = {lo.x, lo.y, lo.z, lo.w, hiq.x, hiq.y, hiq.z, hiq.w};
        rfft8(row, I[r]);
    }
#pragma unroll
    for (int k = 0; k < 5; ++k) {
        cpx v[8];
#pragma unroll
        for (int r = 0; r < 8; ++r) v[r] = I[r][k];
        fft8c<true>(v);
#pragma unroll
        for (int r = 0; r < 8; ++r) I[r][k] = v[r];
    }

    // ---- fuse: (I + 1) * E ----
#pragma unroll
    for (int r = 0; r < 8; ++r)
#pragma unroll
        for (int k = 0; k < 5; ++k) {
            cpx ip1 = {I[r][k].re + 1.0f, I[r][k].im};
            I[r][k] = cmul(ip1, E[r][k]);
        }

    // ---- inverse: columns (unscaled) then rows (1/64 scale) ----
#pragma unroll
    for (int k = 0; k < 5; ++k) {
        cpx v[8];
#pragma unroll
        for (int r = 0; r < 8; ++r) v[r] = I[r][k];
        fft8c<false>(v);
#pragma unroll
        for (int r = 0; r < 8; ++r) I[r][k] = v[r];
    }
#pragma unroll
    for (int r = 0; r < 8; ++r) {
        float o[8];
        irfft8_scaled(I[r], o);
        float4* wp = reinterpret_cast<float4*>(xo + base + (size_t)r * 256);
        wp[0] = make_float4(o[0], o[1], o[2], o[3]);
        wp[1] = make_float4(o[4], o[5], o[6], o[7]);
    }
}

// ---------------------------------------------------------------------------
// Kernel: depthwise 3x3 (SAME, zero pad) + GELU gate.
//   g[b,c,p] = gelu(dw(x[b,c])) * dw(x[b,c+64]),  c in [0,64)
// ---------------------------------------------------------------------------
__device__ __forceinline__ float dw3x3(const float* __restrict__ ch,
                                       const float* __restrict__ w9,
                                       int y, int x) {
    float acc = 0.f;
#pragma unroll
    for (int ky = 0; ky < 3; ++ky) {
        int yy = y + ky - 1;
        if (yy < 0 || yy > 255) continue;
#pragma unroll
        for (int kx = 0; kx < 3; ++kx) {
            int xx = x + kx - 1;
            if (xx < 0 || xx > 255) continue;
            acc = fmaf(ch[yy * 256 + xx], w9[ky * 3 + kx], acc);
        }
    }
    return acc;
}

__global__ __launch_bounds__(256) void dw_gate(const float* __restrict__ x,
                                               const float* __restrict__ wdw,
                                               float* __restrict__ g) {
    int t = blockIdx.x * blockDim.x + threadIdx.x;
    const int p = t & (HWIMG - 1); t >>= 16;
    const int c = t & 63;          t >>= 6;
    const int b = t;
    const int y = p >> 8, xx = p & 255;

    const float* c1 = x + (size_t)(b * HID2 + c) * HWIMG;
    const float* c2 = x + (size_t)(b * HID2 + c + 64) * HWIMG;
    float x1 = dw3x3(c1, wdw + c * 9, y, xx);
    float x2 = dw3x3(c2, wdw + (c + 64) * 9, y, xx);
    float gl = 0.5f * x1 * (1.0f + erff(x1 * 0.70710678118654752f));  // exact gelu
    g[(size_t)(b * 64 + c) * HWIMG + p] = gl * x2;
}

// ---------------------------------------------------------------------------
// Launcher. Workspace layout (needs 256 MB):
//   [0, 128MB)   : img_ (1x1 conv out, HID2 ch)  -> overwritten in place by x
//   [128, 256MB) : event_ (1x1 conv out)         -> reused for g (64 MB)
// ---------------------------------------------------------------------------
extern "C" void kernel_launch(void* const* d_in, const int* in_sizes, int n_in,
                              void* d_out, int out_size, void* d_ws, size_t ws_size,
                              hipStream_t stream) {
    const float* img    = (const float*)d_in[0];
    const float* event_ = (const float*)d_in[1];
    const float* w_img  = (const float*)d_in[2];
    const float* w_evt  = (const float*)d_in[3];
    const float* w_dw   = (const float*)d_in[4];
    const float* filt   = (const float*)d_in[5];
    const float* w_out  = (const float*)d_in[6];
    float*       out    = (float*)d_out;

    const size_t nBig = (size_t)BATCH * HID2 * HWIMG;   // 33.5M floats = 128 MB
    float* xImg = (float*)d_ws;            // img_ then x (in place)
    float* xEvt = (float*)d_ws + nBig;     // event_ then g
    (void)ws_size; (void)in_sizes; (void)n_in; (void)out_size;

    // 1) input 1x1 convs, fp32 WMMA GEMM: OC=128, IC=64, 8 M-tiles x 1 N-tile
    dim3 gA(HWIMG / 16, BATCH);
    conv1x1_wmma<HID2, CIN, 8, 1><<<gA, 256, 0, stream>>>(img,    w_img, xImg);
    conv1x1_wmma<HID2, CIN, 8, 1><<<gA, 256, 0, stream>>>(event_, w_evt, xEvt);

    // 2) patchwise FFT fuse (in place over xImg)
    const int nPatch = BATCH * HID2 * 32 * 32;          // 524288
    fft_fuse<<<nPatch / 256, 256, 0, stream>>>(xImg, xEvt, filt, xImg);

    // 3) depthwise 3x3 + gelu gate -> g (reuse xEvt region)
    const int nG = BATCH * CIN * HWIMG;                 // 16.7M
    dw_gate<<<nG / 256, 256, 0, stream>>>(xImg, w_dw, xEvt);

    // 4) output 1x1 conv, WMMA GEMM: OC=64, IC=64, 4 M-tiles x 2 N-tiles
    dim3 gC(HWIMG / 32, BATCH);
    conv1x1_wmma<CIN, CIN, 4, 2><<<gC, 256, 0, stream>>>(xEvt, w_out, out);
}